// AutoInt_51539607704
// MI455X (gfx1250) — compile-verified
//
#include <hip/hip_runtime.h>

typedef __attribute__((ext_vector_type(2))) float v2f;
typedef __attribute__((ext_vector_type(4))) float v4f;
typedef __attribute__((ext_vector_type(8))) float v8f;

#define N       64
#define LDSS    68          // padded LDS row stride (floats): bank = (4*row+col)%64
#define LAYERS  3

__global__ __launch_bounds__(128) void autoint_attn_kernel(
    const float* __restrict__ in, float* __restrict__ out)
{
  __shared__ float Xs[N * LDSS];   // current activations X
  __shared__ float As[N * LDSS];   // softmax(X X^T) probabilities

  const int b    = blockIdx.x;
  const int tid  = threadIdx.x;
  const int lane = tid & 31;
  const int band = tid >> 5;       // wave id: owns rows [16*band, 16*band+16)
  const int half = lane >> 4;      // 0 = lanes 0-15, 1 = lanes 16-31
  const int l16  = lane & 15;

  const float* gin = in + (size_t)b * (N * N);

  // ---- cooperative coalesced load: 64x64 f32 tile -> LDS (b128) ----
#pragma unroll
  for (int i = 0; i < 8; ++i) {
    int idx = i * 128 + tid;        // 1024 float4 chunks
    int row = idx >> 4;
    int c4  = (idx & 15) << 2;
    v4f v = *(const v4f*)(gin + row * N + c4);
    *(v4f*)(&Xs[row * LDSS + c4]) = v;
  }
  __syncthreads();

  const int arow = band * 16 + l16;   // A-fragment row for this lane
  const int kofs = half * 2;          // K sub-offset per f32 A/B fragment layout
  const float LOG2E = 1.4426950408889634f;

  for (int layer = 0; layer < LAYERS; ++layer) {
    // ================= GEMM1: S = X * X^T (row band, all 64 cols) ==========
    v8f s0 = {}; v8f s1 = {}; v8f s2 = {}; v8f s3 = {};
#pragma unroll
    for (int kb = 0; kb < 16; ++kb) {
      int kcol = kb * 4 + kofs;
      v2f a  = *(const v2f*)(&Xs[arow * LDSS + kcol]);
      // B = X^T  =>  B-frag for N-tile n mirrors the A layout on rows n*16+l16
      v2f b0 = *(const v2f*)(&Xs[( 0 + l16) * LDSS + kcol]);
      v2f b1 = *(const v2f*)(&Xs[(16 + l16) * LDSS + kcol]);
      v2f b2 = *(const v2f*)(&Xs[(32 + l16) * LDSS + kcol]);
      v2f b3 = *(const v2f*)(&Xs[(48 + l16) * LDSS + kcol]);
      s0 = __builtin_amdgcn_wmma_f32_16x16x4_f32(false, a, false, b0, (short)0, s0, false, false);
      s1 = __builtin_amdgcn_wmma_f32_16x16x4_f32(false, a, false, b1, (short)0, s1, false, false);
      s2 = __builtin_amdgcn_wmma_f32_16x16x4_f32(false, a, false, b2, (short)0, s2, false, false);
      s3 = __builtin_amdgcn_wmma_f32_16x16x4_f32(false, a, false, b3, (short)0, s3, false, false);
    }

    // ================= row softmax in C-fragment layout ====================
    // element (vgpr r, lane): row = 16*band + r + 8*half, col = 16*n + l16
#pragma unroll
    for (int r = 0; r < 8; ++r) {
      float m = fmaxf(fmaxf(s0[r], s1[r]), fmaxf(s2[r], s3[r]));
      m = fmaxf(m, __shfl_xor(m, 8, 32));
      m = fmaxf(m, __shfl_xor(m, 4, 32));
      m = fmaxf(m, __shfl_xor(m, 2, 32));
      m = fmaxf(m, __shfl_xor(m, 1, 32));
      float e0 = exp2f((s0[r] - m) * LOG2E);
      float e1 = exp2f((s1[r] - m) * LOG2E);
      float e2 = exp2f((s2[r] - m) * LOG2E);
      float e3 = exp2f((s3[r] - m) * LOG2E);
      float sum = (e0 + e1) + (e2 + e3);
      sum += __shfl_xor(sum, 8, 32);
      sum += __shfl_xor(sum, 4, 32);
      sum += __shfl_xor(sum, 2, 32);
      sum += __shfl_xor(sum, 1, 32);
      float inv = 1.0f / sum;
      int row = band * 16 + r + half * 8;
      As[row * LDSS +  0 + l16] = e0 * inv;
      As[row * LDSS + 16 + l16] = e1 * inv;
      As[row * LDSS + 32 + l16] = e2 * inv;
      As[row * LDSS + 48 + l16] = e3 * inv;
    }
    // As rows of this band are written and read only by this wave: no barrier.

    // ================= GEMM2: Xnew = A * X (row band) ======================
    v8f o0 = {}; v8f o1 = {}; v8f o2 = {}; v8f o3 = {};
#pragma unroll
    for (int kb = 0; kb < 16; ++kb) {
      int krow = kb * 4 + kofs;
      v2f a = *(const v2f*)(&As[arow * LDSS + krow]);
      v2f b0, b1, b2, b3;
      b0.x = Xs[krow * LDSS +  0 + l16]; b0.y = Xs[(krow + 1) * LDSS +  0 + l16];
      b1.x = Xs[krow * LDSS + 16 + l16]; b1.y = Xs[(krow + 1) * LDSS + 16 + l16];
      b2.x = Xs[krow * LDSS + 32 + l16]; b2.y = Xs[(krow + 1) * LDSS + 32 + l16];
      b3.x = Xs[krow * LDSS + 48 + l16]; b3.y = Xs[(krow + 1) * LDSS + 48 + l16];
      o0 = __builtin_amdgcn_wmma_f32_16x16x4_f32(false, a, false, b0, (short)0, o0, false, false);
      o1 = __builtin_amdgcn_wmma_f32_16x16x4_f32(false, a, false, b1, (short)0, o1, false, false);
      o2 = __builtin_amdgcn_wmma_f32_16x16x4_f32(false, a, false, b2, (short)0, o2, false, false);
      o3 = __builtin_amdgcn_wmma_f32_16x16x4_f32(false, a, false, b3, (short)0, o3, false, false);
    }

    __syncthreads();    // everyone done reading Xs before it is overwritten
#pragma unroll
    for (int r = 0; r < 8; ++r) {
      int row = band * 16 + r + half * 8;
      Xs[row * LDSS +  0 + l16] = o0[r];
      Xs[row * LDSS + 16 + l16] = o1[r];
      Xs[row * LDSS + 32 + l16] = o2[r];
      Xs[row * LDSS + 48 + l16] = o3[r];
    }
    __syncthreads();    // Xs consistent for next layer / final store
  }

  // ---- coalesced store: LDS -> global (b128) ----
  float* gout = out + (size_t)b * (N * N);
#pragma unroll
  for (int i = 0; i < 8; ++i) {
    int idx = i * 128 + tid;
    int row = idx >> 4;
    int c4  = (idx & 15) << 2;
    *(v4f*)(gout + row * N + c4) = *(const v4f*)(&Xs[row * LDSS + c4]);
  }
}

extern "C" void kernel_launch(void* const* d_in, const int* in_sizes, int n_in,
                              void* d_out, int out_size, void* d_ws, size_t ws_size,
                              hipStream_t stream) {
  const float* embeds = (const float*)d_in[0];
  float* out = (float*)d_out;
  int B = in_sizes[0] / (N * N);   // 8192
  autoint_attn_kernel<<<B, 128, 0, stream>>>(embeds, out);
}